// LearnablePerChannelProcessor_1176821039743
// MI455X (gfx1250) — compile-verified
//
#include <hip/hip_runtime.h>
#include <hip/hip_bf16.h>

typedef __attribute__((ext_vector_type(2))) float v2f;
typedef __attribute__((ext_vector_type(8))) float v8f;

#define CNUM 256
#define LNUM 720
#define SEG  120          // L positions per block
#define NSEG 6            // 6*120 = 720
#define CG   16           // channels per block (WMMA M)
#define NT   8            // 8 position tiles of 16 cover 126 extended positions
#define XS_W 160          // padded x positions per block (covers [s0-16, s0+144))
#define MS_W 132          // padded ms stride (bank-conflict-free stores)

// -------- Kernel 1: fold softmax(scale_weights), the three depthwise kernels and
// the mean-over-S into one per-channel 5x7 kernel g[c][s'][dw]. ------------------
__global__ __launch_bounds__(64)
void fold_weights_kernel(const float* __restrict__ w3, const float* __restrict__ w5,
                         const float* __restrict__ w7, const float* __restrict__ sw,
                         float* __restrict__ g) {
    int c = blockIdx.x;
    int t = threadIdx.x;
    if (t >= 35) return;
    int sp  = t / 7;        // s' in [0,5)
    int dwi = t % 7;        // dw = dwi - 3
    // softmax over 3 scale weights
    float m  = fmaxf(sw[0], fmaxf(sw[1], sw[2]));
    float e0 = __expf(sw[0] - m), e1 = __expf(sw[1] - m), e2 = __expf(sw[2] - m);
    float inv = 1.0f / (e0 + e1 + e2);
    float wj[3] = {e0 * inv, e1 * inv, e2 * inv};
    const float* K[3] = {w3 + (size_t)c * 9, w5 + (size_t)c * 25, w7 + (size_t)c * 49};
    const int kh[3] = {3, 5, 7};
    float acc = 0.0f;
    for (int j = 0; j < 3; ++j) {
        int chj = kh[j] >> 1;
        int col = dwi - 3 + chj;
        if (col < 0 || col >= kh[j]) continue;
        float s_acc = 0.0f;
        for (int s = 0; s < 5; ++s) {
            int r = sp - s + chj;        // kernel row index
            if (r >= 0 && r < kh[j]) s_acc += K[j][r * kh[j] + col];
        }
        acc += wj[j] * 0.2f * s_acc;     // 0.2 = mean over S
    }
    g[(size_t)c * 35 + t] = acc;
}

// band indicator for window half-width h
#define BAND(d, h) ((((d) >= -(h)) && ((d) <= (h))) ? 1.0f : 0.0f)

// -------- Kernel 2: box sums via v_wmma_f32_16x16x4_f32, edge-corrected means,
// 35-tap per-channel combine, residual blend. ------------------------------------
__global__ __launch_bounds__(256)
void msmean_kernel(const float* __restrict__ x, const float* __restrict__ g,
                   const float* __restrict__ rwp, float* __restrict__ out, int Bn) {
    __shared__ float xs[XS_W * CG];          // zero-padded x tile, [xi][ch]
    __shared__ float msb[5 * CG * MS_W];     // ms staging, [w][ch][pl]

    const int seg = blockIdx.x, cg = blockIdx.y, b = blockIdx.z;
    const int s0 = seg * SEG;
    const int cbase = cg * CG;
    const int tid = threadIdx.x;

    // ---- Phase 1: stage x (zero-padded outside [0,720)) into LDS ----
    for (int t = tid; t < XS_W * CG; t += 256) {
        int ch = t / XS_W;
        int xi = t - ch * XS_W;
        int p  = s0 - 16 + xi;
        float v = 0.0f;
        if (p >= 0 && p < LNUM)
            v = x[((size_t)(b * CNUM + cbase + ch)) * LNUM + p];
        xs[xi * CG + ch] = v;
    }
    __syncthreads();

    // ---- Phase 2: box sums for the 5 windows via WMMA; one 16-pos tile per wave ----
    {
        const int lane = tid & 31;
        const int wv   = tid >> 5;       // wave id = tile id (8 waves, NT=8)
        const int jj   = lane & 15;      // M (A) / N (B) index
        const int hi   = lane >> 4;      // K half select
        const int tt   = wv;

        v8f acc0 = {}, acc1 = {}, acc2 = {}, acc3 = {}, acc4 = {};
        #pragma unroll
        for (int q = 0; q < 10; ++q) {
            // A: x[ch = jj, p = (s0-3+16*tt) - 11 + 4q + k], k = 2*hi (+1 for .y)
            int xi0 = 16 * tt + 2 + 4 * q + 2 * hi;
            v2f av;
            av.x = xs[xi0 * CG + jj];
            av.y = xs[(xi0 + 1) * CG + jj];
            // B: band[p, pos] ; d = p - pos = 4q + k - 11 - jj
            int d0 = 4 * q + 2 * hi - 11 - jj;
            int d1 = d0 + 1;
            v2f bw;
            bw.x = BAND(d0, 7);  bw.y = BAND(d1, 7);
            acc0 = __builtin_amdgcn_wmma_f32_16x16x4_f32(false, av, false, bw, (short)0, acc0, false, false);
            bw.x = BAND(d0, 8);  bw.y = BAND(d1, 8);
            acc1 = __builtin_amdgcn_wmma_f32_16x16x4_f32(false, av, false, bw, (short)0, acc1, false, false);
            bw.x = BAND(d0, 9);  bw.y = BAND(d1, 9);
            acc2 = __builtin_amdgcn_wmma_f32_16x16x4_f32(false, av, false, bw, (short)0, acc2, false, false);
            bw.x = BAND(d0, 10); bw.y = BAND(d1, 10);
            acc3 = __builtin_amdgcn_wmma_f32_16x16x4_f32(false, av, false, bw, (short)0, acc3, false, false);
            bw.x = BAND(d0, 11); bw.y = BAND(d1, 11);
            acc4 = __builtin_amdgcn_wmma_f32_16x16x4_f32(false, av, false, bw, (short)0, acc4, false, false);
        }

        // edge-corrected counts -> means; store ms to LDS
        const int pl  = 16 * tt + jj;           // local ms index
        const int pos = s0 - 3 + pl;            // global position
        const bool valid = (pos >= 0 && pos < LNUM);
        const int hs[5] = {7, 8, 9, 10, 11};
        float invs[5];
        #pragma unroll
        for (int w = 0; w < 5; ++w) {
            float iv = 0.0f;
            if (valid) {
                int lo = pos - hs[w]; if (lo < 0) lo = 0;
                int hp = pos + hs[w]; if (hp > LNUM - 1) hp = LNUM - 1;
                iv = 1.0f / (float)(hp - lo + 1);
            }
            invs[w] = iv;
        }
        #pragma unroll
        for (int v = 0; v < 8; ++v) {
            int ch = v + 8 * hi;                 // D layout: M = v (+8 for hi half)
            msb[(0 * CG + ch) * MS_W + pl] = acc0[v] * invs[0];
            msb[(1 * CG + ch) * MS_W + pl] = acc1[v] * invs[1];
            msb[(2 * CG + ch) * MS_W + pl] = acc2[v] * invs[2];
            msb[(3 * CG + ch) * MS_W + pl] = acc3[v] * invs[3];
            msb[(4 * CG + ch) * MS_W + pl] = acc4[v] * invs[4];
        }
    }
    __syncthreads();

    // ---- Phase 3: 35-tap per-channel combine + residual blend ----
    {
        const int ch = tid >> 4;                 // channel within group
        const int pj = tid & 15;                 // position lane -> coalesced stores
        float garr[35];
        const float* gp = g + (size_t)(cbase + ch) * 35;
        #pragma unroll
        for (int k = 0; k < 35; ++k) garr[k] = gp[k];
        const float rw = rwp[0];
        const float om = 1.0f - rw;

        for (int i = 0; i < 8; ++i) {
            int pos_l = pj + 16 * i;
            if (pos_l < SEG) {
                int pos = s0 + pos_l;
                float acc = 0.0f;
                #pragma unroll
                for (int w = 0; w < 5; ++w) {
                    const float* mrow = &msb[(w * CG + ch) * MS_W + pos_l];
                    #pragma unroll
                    for (int dwi = 0; dwi < 7; ++dwi)
                        acc = fmaf(garr[w * 7 + dwi], mrow[dwi], acc);
                }
                float xv = xs[(pos_l + 16) * CG + ch];
                out[((size_t)(b * CNUM + cbase + ch)) * LNUM + pos] = rw * acc + om * xv;
            }
        }
    }
    (void)Bn;
}

extern "C" void kernel_launch(void* const* d_in, const int* in_sizes, int n_in,
                              void* d_out, int out_size, void* d_ws, size_t ws_size,
                              hipStream_t stream) {
    const float* x  = (const float*)d_in[0];
    const float* w3 = (const float*)d_in[1];
    const float* w5 = (const float*)d_in[2];
    const float* w7 = (const float*)d_in[3];
    const float* sw = (const float*)d_in[4];
    const float* rw = (const float*)d_in[5];
    float* g = (float*)d_ws;                       // 256*35 floats of scratch
    int Bn = in_sizes[0] / (CNUM * LNUM);          // 32

    fold_weights_kernel<<<CNUM, 64, 0, stream>>>(w3, w5, w7, sw, g);
    msmean_kernel<<<dim3(NSEG, CNUM / CG, Bn), 256, 0, stream>>>(x, g, rw, (float*)d_out, Bn);
}